// MCSA_39273180955624
// MI455X (gfx1250) — compile-verified
//
#include <hip/hip_runtime.h>
#include <hip/hip_bf16.h>
#include <cstdint>
#include <cstddef>

#define BT 2
#define CCH 64
#define IMH 128
#define IMW 128
#define NPIX (IMH*IMW)          // 16384
#define NHEADS 8
#define DHEAD 32

typedef __attribute__((ext_vector_type(2))) float v2f;
typedef __attribute__((ext_vector_type(8))) float v8f;

// ---------------------------------------------------------------------------
// Generic WMMA-f32 GEMM: C[M,N] (+)= A[M,K] * B[K,N].
// A addressing:
//   mode 2 (conv9): row -> (b,pixel); k -> (ci = k/9, tap = k%9); 3x3 tap
//                   shift with zero (padMode 0) / reflect (padMode 1) pad.
//                   A element at A[b*sAb + pix'*sAp + ci*sAc].
//                   Requires K = 9*Cin and B = OIHW weight viewed as
//                   [N][9*Cin] row-major (sBk=1, sBn=9*Cin).
//   mode 1 (lin):   A element (m,k) at A[m*sAm + k*sAk].
// B element (k,n) at Bw[k*sBk + n*sBn].
// B tile staged via global_load_async_to_lds_b128 when sBk==1 (unit stride);
// strided B (Gram GEMM) falls back to scalar staging + prefetch.
// Epilogue: optional accumulate, bias[n], activation.
// ---------------------------------------------------------------------------
struct GP {
  const float* A;
  long sAb, sAp, sAc;      // conv9-mode strides (batch, pixel, channel)
  long sAm, sAk;           // lin-mode strides
  const float* Bw;
  long sBk, sBn;
  float* C;
  long ldc;
  const float* bias;       // indexed by n (nullptr -> none); applied iff !accum
  const float* alphaPtr;   // PReLU slope (device scalar)
  int M, N, K;
  int Himg, Wimg;          // conv9-mode image dims
  int padMode;             // 0 zero, 1 reflect
  int mode;                // 2 conv9, 1 lin
  int accum;               // 1: C += tile (read old C)
  int act;                 // 0 none, 1 relu, 2 leaky(0.01), 3 prelu(*alphaPtr)
};

#define TM 128
#define TN 64
#define TKC 16
#define BSTR (TKC + 4)           // Bs row stride (floats); 80B keeps 16B align

__global__ __launch_bounds__(256) void k_gemm_wmma(GP P) {
  __shared__ float As[TKC][TM + 4];   // k-major
  __shared__ float Bs[TN][BSTR];      // n-major, k contiguous (async B128 dst)

  const int tid  = threadIdx.x;
  const int lane = tid & 31;
  const int wave = tid >> 5;
  const int mBlk = blockIdx.y * TM;
  const int nBlk = blockIdx.x * TN;

  // 8 waves: 4 M-subtiles (32 rows) x 2 N-halves (32 cols); 32x32 per wave.
  const int mSub  = (wave & 3) * 32;
  const int nHalf = (wave >> 2) * 32;
  const int lgrp  = lane >> 4;         // K pair select {0,1} vs {2,3}
  const int lm    = lane & 15;

  v8f acc00 = {}, acc01 = {}, acc10 = {}, acc11 = {};

  const bool asyncB = (P.sBk == 1) && ((P.sBn & 3) == 0);
  const unsigned ldsB = (unsigned)(size_t)(&Bs[0][0]);

  for (int k0 = 0; k0 < P.K; k0 += TKC) {
    // ---- stage A tile: TKC x TM (k-major), predicated scalar path ----
    for (int i = tid; i < TKC * TM; i += 256) {
      int kk = i >> 7, mm = i & 127;
      int row = mBlk + mm, k = k0 + kk;
      float v = 0.f;
      if (row < P.M && k < P.K) {
        if (P.mode == 2) {
          int ci  = k / 9;
          int tap = k - ci * 9;
          int b = row / (P.Himg * P.Wimg);
          int p = row - b * (P.Himg * P.Wimg);
          int y = p / P.Wimg + tap / 3 - 1;
          int x = (p % P.Wimg) + tap % 3 - 1;
          bool ok = true;
          if (P.padMode == 0) {
            if (y < 0 || y >= P.Himg || x < 0 || x >= P.Wimg) ok = false;
          } else {
            y = (y < 0) ? -y : ((y >= P.Himg) ? 2 * P.Himg - 2 - y : y);
            x = (x < 0) ? -x : ((x >= P.Wimg) ? 2 * P.Wimg - 2 - x : x);
          }
          if (ok) v = P.A[(long)b * P.sAb + ((long)y * P.Wimg + x) * P.sAp + (long)ci * P.sAc];
        } else {
          v = P.A[(long)row * P.sAm + (long)k * P.sAk];
        }
      }
      As[kk][mm] = v;
    }

    // ---- stage B tile: TKC x TN into n-major LDS ----
    if (asyncB) {
      // one async 16B LDS-DMA per thread: column nn, k-group kg (4 floats)
      int nn = tid & 63, kg = tid >> 6;
      unsigned dst = ldsB + (unsigned)((nn * BSTR + kg * 4) * 4);
      const float* gp = P.Bw + (long)(nBlk + nn) * P.sBn + (k0 + kg * 4);
      unsigned long long ga = (unsigned long long)(size_t)gp;
      asm volatile("global_load_async_to_lds_b128 %0, %1, off"
                   :: "v"(dst), "v"(ga) : "memory");
    } else {
      for (int i = tid; i < TKC * TN; i += 256) {
        int nn = i & 63, kk = i >> 6;
        int col = nBlk + nn, k = k0 + kk;
        float v = 0.f;
        if (col < P.N && k < P.K) v = P.Bw[(long)k * P.sBk + (long)col * P.sBn];
        Bs[nn][kk] = v;
      }
      if (tid < 16 && (k0 + TKC) < P.K)
        __builtin_prefetch(&P.Bw[(long)(k0 + TKC + tid) * P.sBk], 0, 1);
    }

    if (asyncB) asm volatile("s_wait_asynccnt 0x0" ::: "memory");
    __syncthreads();

    // ---- 16 WMMAs over the K-chunk (4 k-steps x 2x2 tiles) ----
#pragma unroll
    for (int kk = 0; kk < TKC; kk += 4) {
      v2f a0, a1, b0, b1;
      a0.x = As[kk + 2 * lgrp][mSub + lm];
      a0.y = As[kk + 2 * lgrp + 1][mSub + lm];
      a1.x = As[kk + 2 * lgrp][mSub + 16 + lm];
      a1.y = As[kk + 2 * lgrp + 1][mSub + 16 + lm];
      b0.x = Bs[nHalf + lm][kk + 2 * lgrp];
      b0.y = Bs[nHalf + lm][kk + 2 * lgrp + 1];
      b1.x = Bs[nHalf + 16 + lm][kk + 2 * lgrp];
      b1.y = Bs[nHalf + 16 + lm][kk + 2 * lgrp + 1];
      acc00 = __builtin_amdgcn_wmma_f32_16x16x4_f32(false, a0, false, b0, (short)0, acc00, false, false);
      acc01 = __builtin_amdgcn_wmma_f32_16x16x4_f32(false, a0, false, b1, (short)0, acc01, false, false);
      acc10 = __builtin_amdgcn_wmma_f32_16x16x4_f32(false, a1, false, b0, (short)0, acc10, false, false);
      acc11 = __builtin_amdgcn_wmma_f32_16x16x4_f32(false, a1, false, b1, (short)0, acc11, false, false);
    }
    __syncthreads();
  }

  float alpha = 0.25f;
  if (P.act == 3 && P.alphaPtr) alpha = *P.alphaPtr;

  // C/D layout: VGPR r -> M = r (lanes 0-15) / r+8 (lanes 16-31); N = lane%16.
#pragma unroll
  for (int mi = 0; mi < 2; ++mi) {
#pragma unroll
    for (int r = 0; r < 8; ++r) {
      int row = mBlk + mSub + mi * 16 + r + 8 * lgrp;
      if (row < P.M) {
#pragma unroll
        for (int ni = 0; ni < 2; ++ni) {
          int col = nBlk + nHalf + ni * 16 + lm;
          if (col < P.N) {
            float v;
            if (mi == 0) v = ni ? acc01[r] : acc00[r];
            else         v = ni ? acc11[r] : acc10[r];
            long ci = (long)row * P.ldc + col;
            if (P.accum)      v += P.C[ci];
            else if (P.bias)  v += P.bias[col];
            if (P.act == 1)      v = v > 0.f ? v : 0.f;
            else if (P.act == 2) v = v > 0.f ? v : 0.01f * v;
            else if (P.act == 3) v = v > 0.f ? v : alpha * v;
            P.C[ci] = v;
          }
        }
      }
    }
  }
}

// ---------------------------------------------------------------------------
// Column reduction over the pixel axis of an NHWC [BT][NPIX][ld] buffer.
// op 0: mean;  op 1: L2 norm = max(sqrt(sum sq), 1e-12)
// ---------------------------------------------------------------------------
__global__ void k_colreduce(const float* in, float* out, int nrows, int ld, int op) {
  int b = blockIdx.x / ld;
  int col = blockIdx.x % ld;
  const float* base = in + (long)b * nrows * ld + col;
  float s = 0.f;
  for (int p = threadIdx.x; p < nrows; p += blockDim.x) {
    float v = base[(long)p * ld];
    s += op ? v * v : v;
  }
  __shared__ float sm[256];
  sm[threadIdx.x] = s;
  __syncthreads();
  for (int st = 128; st > 0; st >>= 1) {
    if (threadIdx.x < st) sm[threadIdx.x] += sm[threadIdx.x + st];
    __syncthreads();
  }
  if (threadIdx.x == 0) {
    float r = sm[0];
    out[blockIdx.x] = op ? fmaxf(sqrtf(r), 1e-12f) : r / (float)nrows;
  }
}

// ---------------------------------------------------------------------------
// QCO tail (per batch): x_ave = sca(mean)+b; histogram stats; 2-layer ReLU MLP.
// ---------------------------------------------------------------------------
__global__ void k_qco_small(const float* mean, const float* sw, const float* sb,
                            const float* t1w, const float* t1b,
                            const float* t2w, const float* t2b, float* att) {
  int b = blockIdx.x, c = threadIdx.x;
  __shared__ float s_xa[64], s_cat[128], s_h1[64], s_red[64], s_mn[64], s_mx[64];

  float xa = sb[c];
  for (int j = 0; j < 64; ++j) xa += sw[c * 64 + j] * mean[b * 64 + j];
  s_xa[c] = xa; s_mn[c] = xa; s_mx[c] = xa;
  __syncthreads();
  for (int st = 32; st > 0; st >>= 1) {
    if (c < st) {
      s_mn[c] = fminf(s_mn[c], s_mn[c + st]);
      s_mx[c] = fmaxf(s_mx[c], s_mx[c + st]);
    }
    __syncthreads();
  }
  float mn = s_mn[0], mx = s_mx[0];
  float range = mx - mn;
  float inter = range / 64.f;
  float ql = (2.f * c + 1.f) / 128.f * range + mn;   // L = 64
  float sta = 0.f;
  for (int j = 0; j < 64; ++j) {
    float q = 1.f - fabsf(ql - s_xa[j]);
    sta += (q > 1.f - inter) ? q : 0.f;
  }
  s_red[c] = sta;
  __syncthreads();
  for (int st = 32; st > 0; st >>= 1) {
    if (c < st) s_red[c] += s_red[c + st];
    __syncthreads();
  }
  float tot = s_red[0];
  s_cat[c] = sta / tot;
  s_cat[64 + c] = xa;
  __syncthreads();
  float h1 = t1b[c];
  for (int j = 0; j < 128; ++j) h1 += t1w[c * 128 + j] * s_cat[j];
  h1 = fmaxf(h1, 0.f);
  s_h1[c] = h1;
  __syncthreads();
  float h2 = t2b[c];
  for (int j = 0; j < 64; ++j) h2 += t2w[c * 64 + j] * s_h1[j];
  att[b * 64 + c] = fmaxf(h2, 0.f);
}

// ---------------------------------------------------------------------------
// Fold L2 normalization + rescale into Gram matrix, then softmax over e (64).
// ---------------------------------------------------------------------------
__global__ void k_attn_softmax(const float* gram, float* attn,
                               const float* nq, const float* nk,
                               const float* rescale, int d1) {
  int row = blockIdx.x;
  int e = threadIdx.x;
  int h = (row / d1) % NHEADS;
  int b = row / (d1 * NHEADS);
  float nqv = nq[row];                                  // [b][NHEADS*d1] == row
  float nkv = nk[(b * NHEADS + h) * (2 * DHEAD) + e];   // [b][512]
  float v = gram[(long)row * 64 + e] / (nqv * nkv) * rescale[h];
  __shared__ float sm[64];
  sm[e] = v;
  __syncthreads();
  for (int st = 32; st > 0; st >>= 1) {
    if (e < st) sm[e] = fmaxf(sm[e], sm[e + st]);
    __syncthreads();
  }
  float mx = sm[0];
  __syncthreads();
  float ex = expf(v - mx);
  sm[e] = ex;
  __syncthreads();
  for (int st = 32; st > 0; st >>= 1) {
    if (e < st) sm[e] += sm[e + st];
    __syncthreads();
  }
  attn[(long)row * 64 + e] = ex / sm[0];
}

// ---------------------------------------------------------------------------
// torch .view(b,c,h,w) raw-reinterpret quirk: build NHWC Hcat[b][p][0..255]
// from proj_l [b][n][64] and proj_h [b][n][192] treated as flat [c*n+p].
// ---------------------------------------------------------------------------
__global__ void k_quirk_cat(const float* pl, const float* ph, float* hcat, long tot) {
  long i = (long)blockIdx.x * blockDim.x + threadIdx.x;
  if (i >= tot) return;
  int c = (int)(i % 256);
  long t = i / 256;
  int p = (int)(t % NPIX);
  int b = (int)(t / NPIX);
  float v;
  if (c < 64) {
    long flat = (long)c * NPIX + p;
    v = pl[(long)b * NPIX * 64 + flat];
  } else {
    long flat = (long)(c - 64) * NPIX + p;
    v = ph[(long)b * NPIX * 192 + flat];
  }
  hcat[i] = v;
}

// out1[b,p,c] (NHWC) = H_add + H_out*att[b,c] + x_nchw[b,c,p]
__global__ void k_out1_combine(const float* hadd, const float* hout,
                               const float* att, const float* x, float* out1, long tot) {
  long i = (long)blockIdx.x * blockDim.x + threadIdx.x;
  if (i >= tot) return;
  int c = (int)(i & 63);
  long t = i >> 6;
  int p = (int)(t % NPIX);
  int b = (int)(t / NPIX);
  out1[i] = hadd[i] + hout[i] * att[b * 64 + c] + x[((long)b * 64 + c) * NPIX + p];
}

// LayerNorm over channel (last dim, 64). grid.x = BT*NPIX rows, block = 64.
__global__ void k_layernorm64(const float* in, const float* g, const float* be, float* out) {
  long row = blockIdx.x;
  int c = threadIdx.x;
  float v = in[row * 64 + c];
  __shared__ float sm[64];
  sm[c] = v;
  __syncthreads();
  for (int st = 32; st > 0; st >>= 1) { if (c < st) sm[c] += sm[c + st]; __syncthreads(); }
  float mu = sm[0] / 64.f;
  __syncthreads();
  float d = v - mu;
  sm[c] = d * d;
  __syncthreads();
  for (int st = 32; st > 0; st >>= 1) { if (c < st) sm[c] += sm[c + st]; __syncthreads(); }
  float var = sm[0] / 64.f;
  out[row * 64 + c] = d * rsqrtf(var + 1e-5f) * g[c] + be[c];
}

__global__ void k_addvec(const float* a, const float* b, float* o, long tot) {
  long i = (long)blockIdx.x * blockDim.x + threadIdx.x;
  if (i < tot) o[i] = a[i] + b[i];
}

// NHWC [b][p][64] -> NCHW d_out
__global__ void k_nhwc_to_nchw(const float* in, float* out, long tot) {
  long i = (long)blockIdx.x * blockDim.x + threadIdx.x;
  if (i >= tot) return;
  int p = (int)(i % NPIX);
  long t = i / NPIX;
  int c = (int)(t % 64);
  int b = (int)(t / 64);
  out[i] = in[((long)b * NPIX + p) * 64 + c];
}

// ---------------------------------------------------------------------------
// Host-side drivers
// ---------------------------------------------------------------------------
static inline void launch_gemm(const GP& P, hipStream_t s) {
  dim3 grid((P.N + TN - 1) / TN, (P.M + TM - 1) / TM);
  k_gemm_wmma<<<grid, 256, 0, s>>>(P);
}

// 3x3 conv as ONE GEMM with taps folded into K (K = 9*Cin, tap-minor so the
// OIHW weight is a contiguous [N][9*Cin] matrix). Output written exactly once.
static void conv3x3(hipStream_t s, const float* A, long sAb, long sAp, long sAc,
                    const float* Wgt, int Cin, int N, float* C, int padMode,
                    const float* bias, int act, const float* alphaPtr) {
  GP P{};
  P.A = A; P.sAb = sAb; P.sAp = sAp; P.sAc = sAc;
  P.Bw = Wgt; P.sBk = 1; P.sBn = (long)9 * Cin;
  P.C = C; P.ldc = N;
  P.bias = bias; P.alphaPtr = alphaPtr;
  P.M = BT * NPIX; P.N = N; P.K = 9 * Cin;
  P.Himg = IMH; P.Wimg = IMW;
  P.padMode = padMode; P.mode = 2; P.accum = 0; P.act = act;
  launch_gemm(P, s);
}

static void gemm_lin(hipStream_t s, const float* A, long sAm, long sAk,
                     const float* Bw, long sBk, long sBn,
                     float* C, long ldc, const float* bias,
                     int M, int N, int K, int act, const float* alphaPtr) {
  GP P{};
  P.A = A; P.sAm = sAm; P.sAk = sAk;
  P.Bw = Bw; P.sBk = sBk; P.sBn = sBn;
  P.C = C; P.ldc = ldc; P.bias = bias; P.alphaPtr = alphaPtr;
  P.M = M; P.N = N; P.K = K;
  P.Himg = IMH; P.Wimg = IMW;
  P.mode = 1; P.padMode = 0; P.accum = 0; P.act = act;
  launch_gemm(P, s);
}

extern "C" void kernel_launch(void* const* d_in, const int* in_sizes, int n_in,
                              void* d_out, int out_size, void* d_ws, size_t ws_size,
                              hipStream_t stream) {
  (void)in_sizes; (void)n_in; (void)out_size; (void)ws_size;
  const float* x        = (const float*)d_in[0];
  const float* w_down   = (const float*)d_in[1];
  const float* w_mid    = (const float*)d_in[2];
  const float* w_up     = (const float*)d_in[3];
  const float* wq_l     = (const float*)d_in[4];
  const float* wk_m     = (const float*)d_in[5];
  const float* wv_m     = (const float*)d_in[6];
  const float* wq_h     = (const float*)d_in[7];
  const float* rescale  = (const float*)d_in[8];
  const float* w_proj_l = (const float*)d_in[9];
  const float* b_proj_l = (const float*)d_in[10];
  const float* w_proj_h = (const float*)d_in[11];
  const float* b_proj_h = (const float*)d_in[12];
  const float* qco_w1   = (const float*)d_in[13];
  const float* qco_sca_w= (const float*)d_in[14];
  const float* qco_sca_b= (const float*)d_in[15];
  const float* qco_t1_w = (const float*)d_in[16];
  const float* qco_t1_b = (const float*)d_in[17];
  const float* qco_t2_w = (const float*)d_in[18];
  const float* qco_t2_b = (const float*)d_in[19];
  const float* rsp_w1   = (const float*)d_in[20];
  const float* rsp_prelu= (const float*)d_in[21];
  const float* rsp_w2   = (const float*)d_in[22];
  const float* c31_w1   = (const float*)d_in[23];
  const float* c31_prelu= (const float*)d_in[24];
  const float* c31_w2   = (const float*)d_in[25];
  const float* c31_w3   = (const float*)d_in[26];
  const float* ln1_g    = (const float*)d_in[27];
  const float* ln1_b    = (const float*)d_in[28];
  const float* ff1_w1   = (const float*)d_in[29];
  const float* ff1_b1   = (const float*)d_in[30];
  const float* ff1_w2   = (const float*)d_in[31];
  const float* ff1_b2   = (const float*)d_in[32];
  const float* ln2_g    = (const float*)d_in[33];
  const float* ln2_b    = (const float*)d_in[34];
  const float* ff2_w1   = (const float*)d_in[35];
  const float* ff2_b1   = (const float*)d_in[36];
  const float* ff2_w2   = (const float*)d_in[37];
  const float* ff2_b2   = (const float*)d_in[38];

  float* Wsp = (float*)d_ws;
  size_t off = 0;
  auto alloc = [&](size_t nfl) { float* p = Wsp + off; off += nfl; return p; };

  const long BN = (long)BT * NPIX;  // 32768 rows

  float* xL   = alloc(BN * 64);     // w_down output, NHWC
  float* xM   = alloc(BN * 128);
  float* xH   = alloc(BN * 192);
  float* q_l  = alloc(BN * 256);
  float* k_mb = alloc(BN * 512);
  float* v_mb = alloc(BN * 512);
  float* q_h  = alloc(BN * 768);
  float* nq_l = alloc(BT * 256);
  float* nk_m = alloc(BT * 512);
  float* nq_h = alloc(BT * 768);
  float* gram_l = alloc((size_t)BT * NHEADS * 32 * 64);
  float* gram_h = alloc((size_t)BT * NHEADS * 96 * 64);
  float* attn_l = alloc((size_t)BT * NHEADS * 32 * 64);
  float* attn_h = alloc((size_t)BT * NHEADS * 96 * 64);
  float* proj_l = alloc(BN * 64);
  float* proj_h = alloc(BN * 192);
  float* Hcat   = alloc(BN * 256);
  float* rsp_t  = alloc(BN * 64);
  float* H_out  = alloc(BN * 64);
  float* qco_x1 = alloc(BN * 64);
  float* qco_m  = alloc(BT * 64);
  float* att_bf = alloc(BT * 64);
  float* c31_t  = alloc(BN * 64);
  float* c31_t2 = alloc(BN * 64);
  float* H_add  = alloc(BN * 64);
  float* out1   = alloc(BN * 64);
  float* ln_t   = alloc(BN * 64);
  float* ff_a   = alloc(BN * 64);
  float* ff_b   = alloc(BN * 64);
  float* out2   = alloc(BN * 64);
  float* out3   = alloc(BN * 64);

  // attention outputs alias the (by then dead) q buffers
  float* xl_att = q_l;   // [b][n][256]
  float* xh_att = q_h;   // [b][n][768]

  // strides for NCHW input x and NHWC activation buffers
  const long x_sAb = (long)CCH * NPIX, x_sAp = 1, x_sAc = NPIX;

  // ---- QKV 3x3 convs (zero pad), NCHW in -> NHWC out, taps folded in K ----
  conv3x3(stream, x, x_sAb, x_sAp, x_sAc, w_down, 64, 64,  xL, 0, nullptr, 0, nullptr);
  conv3x3(stream, x, x_sAb, x_sAp, x_sAc, w_mid,  64, 128, xM, 0, nullptr, 0, nullptr);
  conv3x3(stream, x, x_sAb, x_sAp, x_sAc, w_up,   64, 192, xH, 0, nullptr, 0, nullptr);

  // ---- linear projections (x @ W^T): B[k][n] = W[n*K+k] ----
  gemm_lin(stream, xL, 64, 1,  wq_l, 1, 64,  q_l,  256, nullptr, (int)BN, 256, 64,  0, nullptr);
  gemm_lin(stream, xM, 128, 1, wk_m, 1, 128, k_mb, 512, nullptr, (int)BN, 512, 128, 0, nullptr);
  gemm_lin(stream, xM, 128, 1, wv_m, 1, 128, v_mb, 512, nullptr, (int)BN, 512, 128, 0, nullptr);
  gemm_lin(stream, xH, 192, 1, wq_h, 1, 192, q_h,  768, nullptr, (int)BN, 768, 192, 0, nullptr);

  // ---- column L2 norms over n (normalization axis of _l2n) ----
  k_colreduce<<<BT * 256, 256, 0, stream>>>(q_l,  nq_l, NPIX, 256, 1);
  k_colreduce<<<BT * 512, 256, 0, stream>>>(k_mb, nk_m, NPIX, 512, 1);
  k_colreduce<<<BT * 768, 256, 0, stream>>>(q_h,  nq_h, NPIX, 768, 1);

  // ---- Gram matrices q^T k (K = n = 16384), per (b, head) ----
  for (int b = 0; b < BT; ++b) {
    for (int h = 0; h < NHEADS; ++h) {
      const long qb  = (long)b * NPIX * 256 + h * 32;
      const long qhb = (long)b * NPIX * 768 + h * 96;
      const long kb  = (long)b * NPIX * 512 + h * 64;
      gemm_lin(stream, q_l + qb, 1, 256, k_mb + kb, 512, 1,
               gram_l + (long)(b * NHEADS + h) * 32 * 64, 64, nullptr,
               32, 64, NPIX, 0, nullptr);
      gemm_lin(stream, q_h + qhb, 1, 768, k_mb + kb, 512, 1,
               gram_h + (long)(b * NHEADS + h) * 96 * 64, 64, nullptr,
               96, 64, NPIX, 0, nullptr);
    }
  }

  // ---- normalize + rescale + softmax ----
  k_attn_softmax<<<BT * NHEADS * 32, 64, 0, stream>>>(gram_l, attn_l, nq_l, nk_m, rescale, 32);
  k_attn_softmax<<<BT * NHEADS * 96, 64, 0, stream>>>(gram_h, attn_h, nq_h, nk_m, rescale, 96);

  // ---- x = attn @ v  (per (b,h): out[n,d] = V[n,2DH] @ attn^T[2DH,d]) ----
  for (int b = 0; b < BT; ++b) {
    for (int h = 0; h < NHEADS; ++h) {
      const float* Vb = v_mb + (long)b * NPIX * 512 + h * 64;
      gemm_lin(stream, Vb, 512, 1,
               attn_l + (long)(b * NHEADS + h) * 32 * 64, 1, 64,
               xl_att + (long)b * NPIX * 256 + h * 32, 256, nullptr,
               NPIX, 32, 64, 0, nullptr);
      gemm_lin(stream, Vb, 512, 1,
               attn_h + (long)(b * NHEADS + h) * 96 * 64, 1, 64,
               xh_att + (long)b * NPIX * 768 + h * 96, 768, nullptr,
               NPIX, 96, 64, 0, nullptr);
    }
  }

  // ---- output projections ----
  gemm_lin(stream, xl_att, 256, 1, w_proj_l, 1, 256, proj_l, 64,  b_proj_l, (int)BN, 64,  256, 0, nullptr);
  gemm_lin(stream, xh_att, 768, 1, w_proj_h, 1, 768, proj_h, 192, b_proj_h, (int)BN, 192, 768, 0, nullptr);

  // ---- .view() quirk gather -> Hcat NHWC [b][n][256] ----
  {
    long tot = BN * 256;
    k_quirk_cat<<<(int)((tot + 255) / 256), 256, 0, stream>>>(proj_l, proj_h, Hcat, tot);
  }

  // ---- ReshapeBlock: rconv3(prelu(rconv3(Hcat))) ----
  conv3x3(stream, Hcat, (long)NPIX * 256, 256, 1, rsp_w1, 256, 64, rsp_t, 1, nullptr, 3, rsp_prelu);
  conv3x3(stream, rsp_t, (long)NPIX * 64, 64, 1, rsp_w2, 64, 64, H_out, 1, nullptr, 0, nullptr);

  // ---- QCO on xL ----
  conv3x3(stream, xL, (long)NPIX * 64, 64, 1, qco_w1, 64, 64, qco_x1, 1, nullptr, 0, nullptr);
  k_colreduce<<<BT * 64, 256, 0, stream>>>(qco_x1, qco_m, NPIX, 64, 0);
  k_qco_small<<<BT, 64, 0, stream>>>(qco_m, qco_sca_w, qco_sca_b,
                                     qco_t1_w, qco_t1_b, qco_t2_w, qco_t2_b, att_bf);

  // ---- Conv3_1: 1x1( rconv3( prelu(rconv3(H_out)) ) ) ----
  conv3x3(stream, H_out, (long)NPIX * 64, 64, 1, c31_w1, 64, 64, c31_t, 1, nullptr, 3, c31_prelu);
  conv3x3(stream, c31_t, (long)NPIX * 64, 64, 1, c31_w2, 64, 64, c31_t2, 1, nullptr, 0, nullptr);
  gemm_lin(stream, c31_t2, 64, 1, c31_w3, 1, 64, H_add, 64, nullptr, (int)BN, 64, 64, 0, nullptr);

  // ---- out1 = H_add + H_out * att_before + x ----
  {
    long tot = BN * 64;
    int blks = (int)((tot + 255) / 256);
    k_out1_combine<<<blks, 256, 0, stream>>>(H_add, H_out, att_bf, x, out1, tot);

    // ---- FF block 1: out2 = FF(LN(out1)) + out1 ----
    k_layernorm64<<<(int)BN, 64, 0, stream>>>(out1, ln1_g, ln1_b, ln_t);
    gemm_lin(stream, ln_t, 64, 1, ff1_w1, 1, 64, ff_a, 64, ff1_b1, (int)BN, 64, 64, 2, nullptr);
    gemm_lin(stream, ff_a, 64, 1, ff1_w2, 1, 64, ff_b, 64, ff1_b2, (int)BN, 64, 64, 0, nullptr);
    k_addvec<<<blks, 256, 0, stream>>>(ff_b, out1, out2, tot);

    // ---- FF block 2: out3 = FF(LN(out2)) + out1 ----
    k_layernorm64<<<(int)BN, 64, 0, stream>>>(out2, ln2_g, ln2_b, ln_t);
    gemm_lin(stream, ln_t, 64, 1, ff2_w1, 1, 64, ff_a, 64, ff2_b1, (int)BN, 64, 64, 2, nullptr);
    gemm_lin(stream, ff_a, 64, 1, ff2_w2, 1, 64, ff_b, 64, ff2_b2, (int)BN, 64, 64, 0, nullptr);
    k_addvec<<<blks, 256, 0, stream>>>(ff_b, out1, out3, tot);

    // ---- NHWC -> NCHW final writeback ----
    k_nhwc_to_nchw<<<blks, 256, 0, stream>>>(out3, (float*)d_out, tot);
  }
}